// HashGrid_82987358094120
// MI455X (gfx1250) — compile-verified
//
#include <hip/hip_runtime.h>
#include <stdint.h>

#define TPB 256

typedef float v4f __attribute__((ext_vector_type(4)));

__global__ __launch_bounds__(TPB) void hashgrid_trilerp_kernel(
    const float* __restrict__ x,      // (N,3)
    const float* __restrict__ feat,   // (CAP,520,5)
    const int*   __restrict__ table,  // (32,32,32)
    float* __restrict__ out_f,        // (N,5)
    float* __restrict__ out_df,       // (N,5,3)
    float* __restrict__ out_m,        // (N,)
    int N)
{
    __shared__ float lds[TPB * 15];   // 15 KB, reused across staging phases

    const int  tid  = threadIdx.x;
    const int  base = blockIdx.x * TPB;
    const int  n    = base + tid;
    // Block-uniform: full block AND N%4==0 (so the dfeats base stays 16B aligned)
    const bool full = (base + TPB <= N) && ((N & 3) == 0);

    // ---------------- load coordinates (b128-coalesced via LDS, NT) ---------
    float px = 0.f, py = 0.f, pz = 0.f;
    if (full) {
        const v4f* src = (const v4f*)(x + (size_t)base * 3); // 3072B*blk -> aligned
        v4f* dst = (v4f*)lds;
        for (int i = tid; i < (TPB * 3) / 4; i += TPB) {
            // coordinate stream is read exactly once -> non-temporal load
            dst[i] = __builtin_nontemporal_load(&src[i]);
        }
        __syncthreads();
        px = lds[tid * 3 + 0];
        py = lds[tid * 3 + 1];
        pz = lds[tid * 3 + 2];
        __syncthreads();   // coords consumed; LDS free for reuse
    } else if (n < N) {
        px = x[(size_t)n * 3 + 0];
        py = x[(size_t)n * 3 + 1];
        pz = x[(size_t)n * 3 + 2];
    }

    // ---------------- per-point trilinear gather/accumulate -----------------
    float maskf = 0.f;
    float accf[5];
    float accd[15];
#pragma unroll
    for (int f = 0; f < 5; ++f) accf[f] = 0.f;
#pragma unroll
    for (int t = 0; t < 15; ++t) accd[t] = 0.f;

    if (n < N) {
        // exact IEEE f32 division to match the reference's x / 0.01
        const float ux = px / 0.01f;
        const float uy = py / 0.01f;
        const float uz = pz / 0.01f;
        const float flx = floorf(ux), fly = floorf(uy), flz = floorf(uz);
        const int   xi = (int)flx, yi = (int)fly, zi = (int)flz;
        const float frx = ux - flx, fry = uy - fly, frz = uz - flz;

        // Pass 1: block-table lookups (RT: 128KB table, ~500x reuse, L0/L2-hot)
        bool    ok = true;
        int64_t rowoff[8];
#pragma unroll
        for (int c = 0; c < 8; ++c) {
            const int ci = (c >> 2) & 1, cj = (c >> 1) & 1, ck = c & 1;
            const int gx = xi + ci, gy = yi + cj, gz = zi + ck;
            const int bx = gx >> 3, by = gy >> 3, bz = gz >> 3;
            const bool inb = ((unsigned)bx < 32u) & ((unsigned)by < 32u) &
                             ((unsigned)bz < 32u);
            int bidx = -1;
            if (inb) bidx = table[(bx * 32 + by) * 32 + bz];
            ok = ok && (bidx >= 0);
            const int vidx = ((gx & 7) * 8 + (gy & 7)) * 8 + (gz & 7);
            rowoff[c] = ((int64_t)bidx * 520 + vidx) * 5;
        }

        // Pass 2: gather 8 feature rows (RT: these are the lines we want L2 to
        // keep — 208MB table vs 192MB L2, with cross-point reuse), accumulate.
        if (ok) {
            maskf = 1.f;
            const float wx[2] = {1.f - frx, frx};
            const float wy[2] = {1.f - fry, fry};
            const float wz[2] = {1.f - frz, frz};
#pragma unroll
            for (int c = 0; c < 8; ++c) {
                const int ci = (c >> 2) & 1, cj = (c >> 1) & 1, ck = c & 1;
                const float X = wx[ci], Y = wy[cj], Z = wz[ck];
                const float YZ = Y * Z;
                const float XZ = X * Z;
                const float XY = X * Y;
                const float cw = X * YZ;
                const float cx = ci ? YZ : -YZ;   // S = 2*corner-1
                const float cy = cj ? XZ : -XZ;
                const float cz = ck ? XY : -XY;
                const float* row = feat + rowoff[c];
#pragma unroll
                for (int f = 0; f < 5; ++f) {
                    const float v = row[f];
                    accf[f]          = fmaf(cw, v, accf[f]);
                    accd[f * 3 + 0]  = fmaf(cx, v, accd[f * 3 + 0]);
                    accd[f * 3 + 1]  = fmaf(cy, v, accd[f * 3 + 1]);
                    accd[f * 3 + 2]  = fmaf(cz, v, accd[f * 3 + 2]);
                }
            }
            // m / V with m==1: 1.0f/0.01f rounds to exactly 100.0f (matches JAX)
#pragma unroll
            for (int t = 0; t < 15; ++t) accd[t] *= (1.0f / 0.01f);
        }
    }

    // ---------------- stores (b128-coalesced via LDS, non-temporal) ----------
    // Outputs are streamed (write-once, never re-read by this kernel):
    // TH_NT keeps them from evicting the feature-table lines out of L2.
    if (full) {
        // feats (5/thread) + mask staged together
#pragma unroll
        for (int f = 0; f < 5; ++f) lds[tid * 5 + f] = accf[f];
        lds[TPB * 5 + tid] = maskf;
        __syncthreads();
        {
            v4f*       df4 = (v4f*)(out_f + (size_t)base * 5);  // 5120B*blk
            const v4f* s4  = (const v4f*)lds;
            for (int i = tid; i < (TPB * 5) / 4; i += TPB)
                __builtin_nontemporal_store(s4[i], &df4[i]);
            v4f*       dm4 = (v4f*)(out_m + base);              // 1024B*blk
            const v4f* m4  = (const v4f*)(lds + TPB * 5);
            for (int i = tid; i < TPB / 4; i += TPB)
                __builtin_nontemporal_store(m4[i], &dm4[i]);
        }
        __syncthreads();
        // dfeats (15/thread)
#pragma unroll
        for (int t = 0; t < 15; ++t) lds[tid * 15 + t] = accd[t];
        __syncthreads();
        {
            v4f*       dd4 = (v4f*)(out_df + (size_t)base * 15); // 15360B*blk
            const v4f* s4  = (const v4f*)lds;
            for (int i = tid; i < (TPB * 15) / 4; i += TPB)
                __builtin_nontemporal_store(s4[i], &dd4[i]);
        }
    } else if (n < N) {
#pragma unroll
        for (int f = 0; f < 5; ++f)
            __builtin_nontemporal_store(accf[f], &out_f[(size_t)n * 5 + f]);
#pragma unroll
        for (int t = 0; t < 15; ++t)
            __builtin_nontemporal_store(accd[t], &out_df[(size_t)n * 15 + t]);
        __builtin_nontemporal_store(maskf, &out_m[n]);
    }
}

extern "C" void kernel_launch(void* const* d_in, const int* in_sizes, int n_in,
                              void* d_out, int out_size, void* d_ws, size_t ws_size,
                              hipStream_t stream) {
    const float* x     = (const float*)d_in[0];  // (N,3) f32
    const float* feat  = (const float*)d_in[1];  // (20000,520,5) f32
    const int*   table = (const int*)d_in[2];    // (32,32,32) i32

    const int N = in_sizes[0] / 3;

    float* out    = (float*)d_out;
    float* out_f  = out;                         // N*5
    float* out_df = out + (size_t)N * 5;         // N*15
    float* out_m  = out + (size_t)N * 20;        // N

    const int blocks = (N + TPB - 1) / TPB;
    hipLaunchKernelGGL(hashgrid_trilerp_kernel, dim3(blocks), dim3(TPB), 0, stream,
                       x, feat, table, out_f, out_df, out_m, N);
}